// DeepFM_49151605736166
// MI455X (gfx1250) — compile-verified
//
#include <hip/hip_runtime.h>

#define F_FIELDS 39
#define VOCAB    100000
#define EMB      16
#define BATCH    16384
#define H1DIM    32
#define H2DIM    32
#define DIN      (F_FIELDS * EMB)   // 624

typedef __attribute__((ext_vector_type(2))) float v2f;
typedef __attribute__((ext_vector_type(8))) float v8f;

// ---------------------------------------------------------------------------
// Kernel 1: embedding gather + first-order + FM second-order, stage deep_in.
// Thread-per-sample: each lane reads one 64B cacheline per field (4x float4),
// with one-field lookahead prefetch (global_prefetch_b8).
// ---------------------------------------------------------------------------
__global__ __launch_bounds__(256)
void dfm_gather(const int* __restrict__ Xi, const float* __restrict__ Xv,
                const float* __restrict__ emb1, const float* __restrict__ emb2,
                float* __restrict__ deep_in, float* __restrict__ partial)
{
    const int b = blockIdx.x * blockDim.x + threadIdx.x;
    if (b >= BATCH) return;

    const int*   xi  = Xi + (size_t)b * F_FIELDS;
    const float* xvp = Xv + (size_t)b * F_FIELDS;
    float*       dst = deep_in + (size_t)b * DIN;

    float s[EMB];
#pragma unroll
    for (int d = 0; d < EMB; ++d) s[d] = 0.f;
    float first = 0.f, sqsum = 0.f;

    int idx_next = xi[0];
    for (int f = 0; f < F_FIELDS; ++f) {
        const int    idx = idx_next;
        const size_t off = (size_t)f * VOCAB + (unsigned)idx;
        if (f + 1 < F_FIELDS) {
            idx_next = xi[f + 1];
            const size_t offn = (size_t)(f + 1) * VOCAB + (unsigned)idx_next;
            __builtin_prefetch(emb2 + offn * EMB, 0, 1);   // next row: 64B line
        }
        const float v = xvp[f];
        first += emb1[off] * v;

        const float4* row = (const float4*)(emb2 + off * EMB);
#pragma unroll
        for (int q = 0; q < 4; ++q) {
            float4 e = row[q];
            e.x *= v; e.y *= v; e.z *= v; e.w *= v;
            ((float4*)(dst + f * EMB))[q] = e;
            s[q * 4 + 0] += e.x; s[q * 4 + 1] += e.y;
            s[q * 4 + 2] += e.z; s[q * 4 + 3] += e.w;
            sqsum += e.x * e.x + e.y * e.y + e.z * e.z + e.w * e.w;
        }
    }

    float fm = 0.f;
#pragma unroll
    for (int d = 0; d < EMB; ++d) fm += s[d] * s[d];
    partial[b] = first + 0.5f * (fm - sqsum);
}

// ---------------------------------------------------------------------------
// Kernel 2: GEMM1  h1 = relu(deep_in[16384x624] @ W1[624x32] + b1)
// One wave -> 16 rows x 32 cols via 2 accumulators, V_WMMA_F32_16X16X4_F32.
// ---------------------------------------------------------------------------
__global__ __launch_bounds__(256)
void dfm_gemm1(const float* __restrict__ deep_in, const float* __restrict__ W1,
               const float* __restrict__ b1, float* __restrict__ h1)
{
    const int lane = threadIdx.x & 31;
    const int wave = threadIdx.x >> 5;
    const int tile = blockIdx.x * 8 + wave;          // 1024 row-tiles of 16
    const int col  = lane & 15;
    const int kh   = (lane >> 4) * 2;                // lanes 16-31 hold K+2,K+3
    const float* arow = deep_in + (size_t)(tile * 16 + col) * DIN;

    v8f acc0 = {}; v8f acc1 = {};
#pragma unroll 4
    for (int k = 0; k < DIN; k += 4) {
        v2f a = *(const v2f*)(arow + k + kh);        // A: (M=col, K=k+kh..+1)
        const float* wk = W1 + (size_t)(k + kh) * H1DIM;
        v2f bf0, bf1;                                // B: (K rows, N=col / col+16)
        bf0.x = wk[col];        bf0.y = wk[H1DIM + col];
        bf1.x = wk[16 + col];   bf1.y = wk[H1DIM + 16 + col];
        acc0 = __builtin_amdgcn_wmma_f32_16x16x4_f32(false, a, false, bf0,
                                                     (short)0, acc0, false, false);
        acc1 = __builtin_amdgcn_wmma_f32_16x16x4_f32(false, a, false, bf1,
                                                     (short)0, acc1, false, false);
    }

    const float bias0 = b1[col], bias1 = b1[16 + col];
    const int mbase = tile * 16 + (lane >> 4) * 8;   // D: VGPR r = row r / r+8
#pragma unroll
    for (int r = 0; r < 8; ++r) {
        float v0 = acc0[r] + bias0; v0 = v0 > 0.f ? v0 : 0.f;
        float v1 = acc1[r] + bias1; v1 = v1 > 0.f ? v1 : 0.f;
        h1[(size_t)(mbase + r) * H1DIM + col]      = v0;
        h1[(size_t)(mbase + r) * H1DIM + 16 + col] = v1;
    }
}

// ---------------------------------------------------------------------------
// Kernel 3: GEMM2 h2 = relu(h1 @ W2 + b2), then per-row sum (shfl_xor within
// each 16-lane half) + partial (first + FM) + global bias -> d_out.
// ---------------------------------------------------------------------------
__global__ __launch_bounds__(256)
void dfm_gemm2(const float* __restrict__ h1, const float* __restrict__ W2,
               const float* __restrict__ b2, const float* __restrict__ partial,
               const float* __restrict__ bias, float* __restrict__ out)
{
    const int lane = threadIdx.x & 31;
    const int wave = threadIdx.x >> 5;
    const int tile = blockIdx.x * 8 + wave;
    const int col  = lane & 15;
    const int kh   = (lane >> 4) * 2;
    const float* arow = h1 + (size_t)(tile * 16 + col) * H1DIM;

    v8f acc0 = {}; v8f acc1 = {};
#pragma unroll
    for (int k = 0; k < H1DIM; k += 4) {
        v2f a = *(const v2f*)(arow + k + kh);
        const float* wk = W2 + (size_t)(k + kh) * H2DIM;
        v2f bf0, bf1;
        bf0.x = wk[col];        bf0.y = wk[H2DIM + col];
        bf1.x = wk[16 + col];   bf1.y = wk[H2DIM + 16 + col];
        acc0 = __builtin_amdgcn_wmma_f32_16x16x4_f32(false, a, false, bf0,
                                                     (short)0, acc0, false, false);
        acc1 = __builtin_amdgcn_wmma_f32_16x16x4_f32(false, a, false, bf1,
                                                     (short)0, acc1, false, false);
    }

    const float bias0 = b2[col], bias1 = b2[16 + col];
    float rowv[8];
#pragma unroll
    for (int r = 0; r < 8; ++r) {
        float v0 = acc0[r] + bias0; v0 = v0 > 0.f ? v0 : 0.f;
        float v1 = acc1[r] + bias1; v1 = v1 > 0.f ? v1 : 0.f;
        rowv[r] = v0 + v1;                 // this lane's two columns of row r
    }
    // Reduce across the 16 lanes of each half-wave (xor masks 1,2,4,8 never
    // cross bit 4, so the two row groups stay independent).
#pragma unroll
    for (int m = 1; m < 16; m <<= 1)
#pragma unroll
        for (int r = 0; r < 8; ++r)
            rowv[r] += __shfl_xor(rowv[r], m, 32);

    if (col == 0) {
        const int mbase = tile * 16 + (lane >> 4) * 8;
        const float bs = bias[0];
#pragma unroll
        for (int r = 0; r < 8; ++r)
            out[mbase + r] = rowv[r] + partial[mbase + r] + bs;
    }
}

// ---------------------------------------------------------------------------
extern "C" void kernel_launch(void* const* d_in, const int* in_sizes, int n_in,
                              void* d_out, int out_size, void* d_ws, size_t ws_size,
                              hipStream_t stream)
{
    const int*   Xi   = (const int*)  d_in[0];
    const float* Xv   = (const float*)d_in[1];
    const float* emb1 = (const float*)d_in[2];
    const float* emb2 = (const float*)d_in[3];
    const float* W1   = (const float*)d_in[4];
    const float* b1   = (const float*)d_in[5];
    const float* W2   = (const float*)d_in[6];
    const float* b2   = (const float*)d_in[7];
    const float* bias = (const float*)d_in[8];
    float* out = (float*)d_out;

    // Workspace layout (all 16B-aligned offsets):
    //   partial : 16384 f32            (64 KB)
    //   h1      : 16384 x 32 f32       ( 2 MB)
    //   deep_in : 16384 x 624 f32      (~41 MB)  -- lives in the 192MB L2
    char*  ws      = (char*)d_ws;
    float* partial = (float*)ws;
    float* h1buf   = (float*)(ws + (64u << 10));
    float* deepin  = (float*)(ws + (64u << 10) + (2u << 20));

    dfm_gather<<<BATCH / 256, 256, 0, stream>>>(Xi, Xv, emb1, emb2, deepin, partial);
    dfm_gemm1 <<<(BATCH / 16) / 8, 256, 0, stream>>>(deepin, W1, b1, h1buf);
    dfm_gemm2 <<<(BATCH / 16) / 8, 256, 0, stream>>>(h1buf, W2, b2, partial, bias, out);
}